// Quantization_85882166051597
// MI455X (gfx1250) — compile-verified
//
#include <hip/hip_runtime.h>

// Quantization (VQ-VAE encode + nearest-code lookup), fused single kernel.
//   z = x @ W^T + b            (32768x512x512 f32 GEMM, v_wmma_f32_16x16x4_f32)
//   d2 argmin == argmin_c ( ||c||^2 - 2 * z.c )   (32768x512x128 GEMM via WMMA)
//   q = codebook[argmin]
// Outputs concatenated: [q (B*512)] [q (B*512)] [z (B*512)]  (all f32)

#define DIM      512
#define NCODES   128
#define LDSTRIDE 516   // 512 + 4 pad: phase-2 ds_load_b64 hits all 64 banks

typedef float v2f __attribute__((ext_vector_type(2)));
typedef float v8f __attribute__((ext_vector_type(8)));

__global__ __launch_bounds__(256) void vq_fused_kernel(
    const float* __restrict__ x,      // [B, 512]
    const float* __restrict__ W,      // [512, 512]  (nn.Linear weight: z = x W^T)
    const float* __restrict__ bvec,   // [512]
    const float* __restrict__ cb,     // [128, 512]
    float* __restrict__ out_q0,
    float* __restrict__ out_q1,
    float* __restrict__ out_z)
{
    __shared__ float lds_z[16 * LDSTRIDE];      // 33 KB  z strip (bias included)
    __shared__ float lds_score[16 * NCODES];    // 8 KB   per-row scores
    __shared__ float lds_cnorm[NCODES];
    __shared__ int   lds_idx[16];

    const int tid   = threadIdx.x;
    const int wave  = tid >> 5;
    const int lane  = tid & 31;
    const int row   = lane & 15;          // fragment row (A) / col (B) index
    const int khalf = (lane >> 4) << 1;   // lanes 0-15 -> K{0,1}, lanes 16-31 -> K{2,3}
    const int hi    = (lane >> 4) << 3;   // C/D: lanes 16-31 hold M = j + 8
    const int col   = lane & 15;          // C/D: N = lane % 16
    const int m0    = blockIdx.x * 16;

    // Warm the codebook rows this wave needs in phase 2 (global_prefetch_b8).
    {
        const float* p = cb + (wave * 16 + row) * DIM;
        __builtin_prefetch(p, 0, 0);
        __builtin_prefetch(p + 256, 0, 0);
    }

    // ---------------- Phase 1: z[16 x 64] strip per wave --------------------
    const int n0 = wave * 64;
    v8f acc[4] = {v8f{}, v8f{}, v8f{}, v8f{}};
    {
        const float* xp  = x + (size_t)(m0 + row) * DIM + khalf;
        const float* wp0 = W + (size_t)(n0 +  0 + row) * DIM + khalf;
        const float* wp1 = W + (size_t)(n0 + 16 + row) * DIM + khalf;
        const float* wp2 = W + (size_t)(n0 + 32 + row) * DIM + khalf;
        const float* wp3 = W + (size_t)(n0 + 48 + row) * DIM + khalf;
        #pragma unroll 4
        for (int k0 = 0; k0 < DIM; k0 += 4) {
            v2f a  = *(const v2f*)(xp  + k0);
            v2f b0 = *(const v2f*)(wp0 + k0);
            v2f b1 = *(const v2f*)(wp1 + k0);
            v2f b2 = *(const v2f*)(wp2 + k0);
            v2f b3 = *(const v2f*)(wp3 + k0);
            acc[0] = __builtin_amdgcn_wmma_f32_16x16x4_f32(false, a, false, b0, (short)0, acc[0], false, false);
            acc[1] = __builtin_amdgcn_wmma_f32_16x16x4_f32(false, a, false, b1, (short)0, acc[1], false, false);
            acc[2] = __builtin_amdgcn_wmma_f32_16x16x4_f32(false, a, false, b2, (short)0, acc[2], false, false);
            acc[3] = __builtin_amdgcn_wmma_f32_16x16x4_f32(false, a, false, b3, (short)0, acc[3], false, false);
        }
    }

    // Spill z strip (+bias) to LDS in C-layout order.
    #pragma unroll
    for (int jj = 0; jj < 4; ++jj) {
        const int n    = n0 + jj * 16 + col;
        const float bb = bvec[n];
        #pragma unroll
        for (int j = 0; j < 8; ++j) {
            lds_z[(j + hi) * LDSTRIDE + n] = acc[jj][j] + bb;
        }
    }

    // ||c||^2 for all 128 codes (codebook is L2-hot).
    if (tid < NCODES) {
        const float4* c4 = (const float4*)(cb + (size_t)tid * DIM);
        float s = 0.f;
        #pragma unroll 8
        for (int i = 0; i < DIM / 4; ++i) {
            float4 v = c4[i];
            s += v.x * v.x + v.y * v.y + v.z * v.z + v.w * v.w;
        }
        lds_cnorm[tid] = s;
    }
    __syncthreads();

    // ---------------- Phase 2: G[16 x 16] per wave (codes 16w..16w+15) ------
    {
        const int c0 = wave * 16;
        v8f g = {};
        const float* cbp = cb + (size_t)(c0 + row) * DIM + khalf;   // B fragment (global)
        const float* zl  = lds_z + row * LDSTRIDE + khalf;          // A fragment (LDS)
        #pragma unroll 4
        for (int k0 = 0; k0 < DIM; k0 += 4) {
            v2f a  = *(const v2f*)(zl  + k0);      // ds_load_b64, conflict-free
            v2f bb = *(const v2f*)(cbp + k0);
            g = __builtin_amdgcn_wmma_f32_16x16x4_f32(false, a, false, bb, (short)0, g, false, false);
        }
        const float cn = lds_cnorm[c0 + col];
        #pragma unroll
        for (int j = 0; j < 8; ++j) {
            lds_score[(j + hi) * NCODES + c0 + col] = cn - 2.0f * g[j];
        }
    }
    __syncthreads();

    // Argmin per row (first-occurrence semantics like jnp.argmin).
    if (tid < 16) {
        const float* s = lds_score + tid * NCODES;
        float best = s[0];
        int   bi   = 0;
        for (int c = 1; c < NCODES; ++c) {
            float v = s[c];
            if (v < best) { best = v; bi = c; }
        }
        lds_idx[tid] = bi;
    }
    __syncthreads();

    // ---------------- Outputs: q (twice) + z, coalesced float4 --------------
    {
        float4*       zo  = (float4*)(out_z  + (size_t)m0 * DIM);
        float4*       q0  = (float4*)(out_q0 + (size_t)m0 * DIM);
        float4*       q1  = (float4*)(out_q1 + (size_t)m0 * DIM);
        const float4* cb4 = (const float4*)cb;
        #pragma unroll
        for (int i = 0; i < 8; ++i) {
            const int f = tid + 256 * i;      // 0..2047 = 16 rows * 128 float4
            const int m = f >> 7;
            const int v = f & 127;
            float4 zv = *(const float4*)(&lds_z[m * LDSTRIDE + 4 * v]);
            zo[m * 128 + v] = zv;
            const int idx = lds_idx[m];
            float4 qv = cb4[(size_t)idx * 128 + v];
            q0[m * 128 + v] = qv;
            q1[m * 128 + v] = qv;
        }
    }
}

extern "C" void kernel_launch(void* const* d_in, const int* in_sizes, int n_in,
                              void* d_out, int out_size, void* d_ws, size_t ws_size,
                              hipStream_t stream) {
    const float* x  = (const float*)d_in[0];
    const float* W  = (const float*)d_in[1];
    const float* b  = (const float*)d_in[2];
    const float* cb = (const float*)d_in[3];

    const int rows = in_sizes[0] / DIM;          // 8*4096 = 32768
    float* out_q0 = (float*)d_out;               // q reshaped (same flat data)
    float* out_q1 = out_q0 + (size_t)rows * DIM; // q
    float* out_z  = out_q1 + (size_t)rows * DIM; // z

    dim3 grid(rows / 16), block(256);
    vq_fused_kernel<<<grid, block, 0, stream>>>(x, W, b, cb, out_q0, out_q1, out_z);
}